// HypAggAttsparse_87582973100272
// MI455X (gfx1250) — compile-verified
//
#include <hip/hip_runtime.h>
#include <hip/hip_bf16.h>
#include <math.h>

#define D 64
#define HEADS 4

typedef float v2f __attribute__((ext_vector_type(2)));
typedef float v8f __attribute__((ext_vector_type(8)));

__device__ __forceinline__ float wave_sum32(float v) {
#pragma unroll
  for (int m = 16; m >= 1; m >>= 1) v += __shfl_xor(v, m, 32);
  return v;
}

__device__ __forceinline__ void atomic_add_f32(float* p, float v) {
  __hip_atomic_fetch_add(p, v, __ATOMIC_RELAXED, __HIP_MEMORY_SCOPE_AGENT);
}

// ---------------------------------------------------------------------------
// zero-fill
__global__ void zero_kernel(float* __restrict__ p, long n) {
  long i = (long)blockIdx.x * blockDim.x + threadIdx.x;
  if (i < n) p[i] = 0.0f;
}

// ---------------------------------------------------------------------------
// x_tan = logmap0(x, c=1): one wave (32 lanes) per node row of 64 floats
__global__ void logmap_kernel(const float* __restrict__ x, float* __restrict__ xt, int N) {
  long wv = ((long)blockIdx.x * blockDim.x + threadIdx.x) >> 5;
  int lane = threadIdx.x & 31;
  if (wv >= N) return;
  const float* row = x + wv * D;
  float v0 = row[lane];
  float v1 = row[lane + 32];
  float ss = wave_sum32(v0 * v0 + v1 * v1);
  float norm = sqrtf(ss);
  float xn = fmaxf(norm, 1e-15f);              // p_norm clamp
  float r = fminf(xn, 1.0f - 1e-7f);           // artanh clip (norm >= 0)
  float at = 0.5f * (log1pf(r) - log1pf(-r));  // artanh
  float sc = at / xn;
  float* o = xt + wv * D;
  o[lane] = v0 * sc;
  o[lane + 32] = v1 * sc;
}

// ---------------------------------------------------------------------------
// h[n, hd*64 + c] = (x_tan @ W[hd])  via V_WMMA_F32_16X16X4_F32
// also s_src[n,hd] = h . a[hd,:64], s_dst[n,hd] = h . a[hd,64:]
// block = 128 threads = 4 waves; wave w owns rows [blk*64 + w*16, +16)
// grid = (ceil(N/64), HEADS)
__global__ __launch_bounds__(128) void wmma_h_kernel(
    const float* __restrict__ xt, const float* __restrict__ W,
    const float* __restrict__ a, float* __restrict__ h,
    float* __restrict__ s, int N) {
  __shared__ float lxt[64][65];  // padded: stride 65 avoids bank conflicts
  __shared__ float lw[64][65];
  const int tid = threadIdx.x;
  const int wave = tid >> 5;
  const int lane = tid & 31;
  const int hd = blockIdx.y;
  const long rbase = (long)blockIdx.x * 64;

  // cooperative load of x_tan tile (clamped rows keep EXEC full for WMMA)
  for (int i = tid; i < 64 * 64; i += 128) {
    int r = i >> 6, c = i & 63;
    long gr = rbase + r;
    if (gr >= N) gr = N - 1;
    lxt[r][c] = xt[gr * D + c];
  }
  const float* Wh = W + (size_t)hd * D * D;
  for (int i = tid; i < 64 * 64; i += 128) {
    int r = i >> 6, c = i & 63;
    lw[r][c] = Wh[i];
  }
  __syncthreads();

  const int mrow = lane & 15;  // A: matrix row M / B: matrix col N
  const int g = lane >> 4;     // lane group -> K offset +2 (A/B), M offset +8 (C/D)
  const int arow = wave * 16 + mrow;

  v8f acc[4] = {};  // 4 column tiles of 16 -> 64 output cols
#pragma unroll
  for (int ks = 0; ks < 16; ks++) {
    const int k0 = ks * 4 + 2 * g;
    v2f af;
    af[0] = lxt[arow][k0];
    af[1] = lxt[arow][k0 + 1];
#pragma unroll
    for (int nt = 0; nt < 4; nt++) {
      v2f bf;
      bf[0] = lw[k0][nt * 16 + mrow];
      bf[1] = lw[k0 + 1][nt * 16 + mrow];
      acc[nt] = __builtin_amdgcn_wmma_f32_16x16x4_f32(
          false, af, false, bf, (short)0, acc[nt], false, false);
    }
  }

  // store h: layout [n][HEADS*D]
#pragma unroll
  for (int nt = 0; nt < 4; nt++) {
#pragma unroll
    for (int v = 0; v < 8; v++) {
      long gr = rbase + wave * 16 + v + 8 * g;
      if (gr < N) h[gr * (HEADS * D) + hd * D + nt * 16 + mrow] = acc[nt][v];
    }
  }

  // fused attention-vector dots; reduce over 16 lanes (stays in lane group)
  const float* al = a + hd * 2 * D;
  const float* ar = al + D;
#pragma unroll
  for (int v = 0; v < 8; v++) {
    float ps = 0.0f, pd = 0.0f;
#pragma unroll
    for (int nt = 0; nt < 4; nt++) {
      float hv = acc[nt][v];
      ps += hv * al[nt * 16 + mrow];
      pd += hv * ar[nt * 16 + mrow];
    }
#pragma unroll
    for (int m = 8; m >= 1; m >>= 1) {
      ps += __shfl_xor(ps, m, 32);
      pd += __shfl_xor(pd, m, 32);
    }
    long gr = rbase + wave * 16 + v + 8 * g;
    if (mrow == 0 && gr < N) {
      s[gr * 2 * HEADS + hd] = ps;            // s_src
      s[gr * 2 * HEADS + HEADS + hd] = pd;    // s_dst
    }
  }
}

// ---------------------------------------------------------------------------
// per-edge: w_hd = exp(-leakyrelu(s_src[src,hd]+s_dst[dst,hd], 0.2))
// num[src] += w_hd * h[dst] ; denom[src,hd] += w_hd     (one wave per edge)
__global__ __launch_bounds__(256) void edge_kernel(
    const int* __restrict__ ei, const float* __restrict__ h,
    const float* __restrict__ s, float* __restrict__ num,
    float* __restrict__ denom, long E) {
  long wv = ((long)blockIdx.x * blockDim.x + threadIdx.x) >> 5;
  int lane = threadIdx.x & 31;
  if (wv >= E) return;
  int src = ei[wv];
  int dst = ei[E + wv];
  float w[HEADS];
#pragma unroll
  for (int hd = 0; hd < HEADS; hd++) {
    float score = s[(long)src * 2 * HEADS + hd] + s[(long)dst * 2 * HEADS + HEADS + hd];
    float lr = score > 0.0f ? score : 0.2f * score;
    w[hd] = expf(-lr);
  }
  const float* hrow = h + (long)dst * (HEADS * D);
  float* nrow = num + (long)src * (HEADS * D);
#pragma unroll
  for (int j = 0; j < 8; j++) {
    int idx = j * 32 + lane;
    atomic_add_f32(&nrow[idx], w[idx >> 6] * hrow[idx]);
  }
  if (lane < HEADS) atomic_add_f32(&denom[(long)src * HEADS + lane], w[lane]);
}

// ---------------------------------------------------------------------------
// h' = num/(denom+eps); leaky(0.01); expmap0 + proj  — in-place on d_out
__global__ void finalize_kernel(float* __restrict__ out, const float* __restrict__ denom, int N) {
  long wv = ((long)blockIdx.x * blockDim.x + threadIdx.x) >> 5;
  int lane = threadIdx.x & 31;
  if (wv >= N) return;
  float* row = out + wv * (HEADS * D);
  float v[8];
  float ss = 0.0f;
#pragma unroll
  for (int j = 0; j < 8; j++) {
    int idx = j * 32 + lane;
    float hp = row[idx] / (denom[wv * HEADS + (idx >> 6)] + 1e-15f);
    hp = hp > 0.0f ? hp : 0.01f * hp;
    v[j] = hp;
    ss += hp * hp;
  }
  ss = wave_sum32(ss);
  float un = fmaxf(sqrtf(ss), 1e-15f);
  float t = tanhf(un);           // resulting norm after expmap0 (c=1)
  float sc = t / un;
  float yn = fmaxf(t, 1e-15f);
  if (yn > 0.996f) sc *= 0.996f / yn;  // proj: maxnorm = (1 - 4e-3)
#pragma unroll
  for (int j = 0; j < 8; j++) row[j * 32 + lane] = v[j] * sc;
}

// ---------------------------------------------------------------------------
extern "C" void kernel_launch(void* const* d_in, const int* in_sizes, int n_in,
                              void* d_out, int out_size, void* d_ws, size_t ws_size,
                              hipStream_t stream) {
  const float* x = (const float*)d_in[0];
  const int* ei = (const int*)d_in[1];
  const float* W = (const float*)d_in[2];
  const float* a = (const float*)d_in[3];
  float* out = (float*)d_out;  // doubles as the `num` accumulator

  int N = in_sizes[0] / D;
  long E = (long)in_sizes[1] / 2;

  float* ws = (float*)d_ws;
  float* xt = ws;                            // N*64
  float* s = xt + (size_t)N * D;             // N*8
  float* denom = s + (size_t)N * 2 * HEADS;  // N*4
  float* h = denom + (size_t)N * HEADS;      // N*256

  long zn = (long)N * HEADS * D;
  zero_kernel<<<(int)((zn + 255) / 256), 256, 0, stream>>>(out, zn);
  long zd = (long)N * HEADS;
  zero_kernel<<<(int)((zd + 255) / 256), 256, 0, stream>>>(denom, zd);

  logmap_kernel<<<(int)(((long)N * 32 + 255) / 256), 256, 0, stream>>>(x, xt, N);

  dim3 gh((N + 63) / 64, HEADS);
  wmma_h_kernel<<<gh, 128, 0, stream>>>(xt, W, a, h, s, N);

  edge_kernel<<<(int)((E * 32 + 255) / 256), 256, 0, stream>>>(ei, h, s, out, denom, E);

  finalize_kernel<<<(int)(((long)N * 32 + 255) / 256), 256, 0, stream>>>(out, denom, N);
}